// NUFFTAdjoint_3917010174181
// MI455X (gfx1250) — compile-verified
//
#include <hip/hip_runtime.h>
#include <math.h>

#define N_IMG   256
#define G_DIM   512
#define NCOIL   12
#define PI_F    3.14159265358979323846f
#define ALPHA_F (2.34f * 6.0f)      // 14.04
#define NSHIFT_F 128.0f
#define INV_SQRT_G 0.04419417382415922f  // 1/sqrt(512)

typedef __attribute__((ext_vector_type(2))) float v2f;
typedef __attribute__((ext_vector_type(8))) float v8f;

// ---------------------------------------------------------------- i0 (A&S 9.8)
__device__ __forceinline__ float bessel_i0f(float x) {
    float ax = fabsf(x);
    if (ax < 3.75f) {
        float t = x / 3.75f; t *= t;
        return 1.0f + t*(3.5156229f + t*(3.0899424f + t*(1.2067492f +
                     t*(0.2659732f + t*(0.0360768f + t*0.0045813f)))));
    } else {
        float t = 3.75f / ax;
        float p = 0.39894228f + t*(0.01328592f + t*(0.00225319f + t*(-0.00157565f +
                  t*(0.00916281f + t*(-0.02057706f + t*(0.02635537f +
                  t*(-0.01647633f + t*0.00392377f)))))));
        return expf(ax) * p / sqrtf(ax);
    }
}

__device__ __forceinline__ float kb_ft(float f) {
    float t  = PI_F * 6.0f * f;
    float z2 = ALPHA_F * ALPHA_F - t * t;
    float z  = sqrtf(fmaxf(z2, 1e-12f));
    return 6.0f * sinhf(z) / z;
}

// ---------------------------------------------------------------- zero scratch
__global__ void zero_f32(float* __restrict__ p, int n) {
    int i = blockIdx.x * blockDim.x + threadIdx.x;
    int stride = gridDim.x * blockDim.x;
    for (; i < n; i += stride) p[i] = 0.0f;
}

// ---------------------------------------------------------------- DFT matrix W
// W[m,n] = exp(+2*pi*i*m*n/G) / sqrt(G)  (split planes; W is symmetric)
__global__ void fill_W(float* __restrict__ Wre, float* __restrict__ Wim) {
    int idx = blockIdx.x * blockDim.x + threadIdx.x;
    if (idx >= G_DIM * G_DIM) return;
    int m = idx >> 9, n = idx & (G_DIM - 1);
    int p = (m * n) & (G_DIM - 1);          // exact integer phase
    float ang = (float)p * (2.0f * PI_F / (float)G_DIM);
    float s, c;
    __sincosf(ang, &s, &c);
    Wre[idx] = c * INV_SQRT_G;
    Wim[idx] = s * INV_SQRT_G;
}

// ---------------------------------------------------------------- KB gridding
__global__ void gridding(const float* __restrict__ inp,    // (C,K,2)
                         const float* __restrict__ ktraj,  // (2,K)
                         const float* __restrict__ dcomp,  // (K)
                         float* __restrict__ gre,          // (C,G,G)
                         float* __restrict__ gim,          // (C,G,G)
                         int K) {
    int k = blockIdx.x * blockDim.x + threadIdx.x;
    if (k >= K) return;

    float kx = ktraj[k], ky = ktraj[K + k];
    float dc = dcomp[k];
    float sp, cp;
    sincosf(NSHIFT_F * (kx + ky), &sp, &cp);
    float sre = dc * cp, sim = dc * sp;

    const float sc = (float)G_DIM / (2.0f * PI_F);
    float lx = fmodf(kx * sc, (float)G_DIM); if (lx < 0.0f) lx += (float)G_DIM;
    float ly = fmodf(ky * sc, (float)G_DIM); if (ly < 0.0f) ly += (float)G_DIM;
    float flx = floorf(lx), fly = floorf(ly);
    float frx = lx - flx,  fry = ly - fly;
    int bx = (int)flx, by = (int)fly;

    float wx[6], wy[6];
    int   ix[6], iy[6];
#pragma unroll
    for (int j = 0; j < 6; ++j) {
        int off = j - 2;                         // offs = -2..3
        float dx = frx - (float)off;
        float xx = dx * (1.0f / 3.0f);
        wx[j] = bessel_i0f(ALPHA_F * sqrtf(fmaxf(1.0f - xx * xx, 0.0f)));
        float dy = fry - (float)off;
        float yy = dy * (1.0f / 3.0f);
        wy[j] = bessel_i0f(ALPHA_F * sqrtf(fmaxf(1.0f - yy * yy, 0.0f)));
        int vx = bx + off; vx += (vx < 0) ? G_DIM : 0; vx -= (vx >= G_DIM) ? G_DIM : 0;
        int vy = by + off; vy += (vy < 0) ? G_DIM : 0; vy -= (vy >= G_DIM) ? G_DIM : 0;
        ix[j] = vx; iy[j] = vy;
    }

    for (int c = 0; c < NCOIL; ++c) {
        float ar = inp[((size_t)c * K + k) * 2];
        float ai = inp[((size_t)c * K + k) * 2 + 1];
        float vr = ar * sre - ai * sim;
        float vi = ar * sim + ai * sre;
        size_t cb = (size_t)c * G_DIM * G_DIM;
#pragma unroll
        for (int jx = 0; jx < 6; ++jx) {
            size_t rowb = cb + (size_t)ix[jx] * G_DIM;
#pragma unroll
            for (int jy = 0; jy < 6; ++jy) {
                float w = wx[jx] * wy[jy];
                size_t cell = rowb + iy[jy];
                atomicAdd(gre + cell, vr * w);
                atomicAdd(gim + cell, vi * w);
            }
        }
    }
}

// ---------------------------------------------------------------- complex GEMM
// D[c] = A[c] x B[c], all split-plane fp32, via V_WMMA_F32_16X16X4_F32.
//   A: row-major,  [row*lda + k]     (K contiguous)
//   B: K-major,    [col*ldb + k]     (K contiguous; pass W symmetric / T_t)
//   D: TRANSPOSED, [col*M + m]       (each lane stores 8 contiguous floats)
// One wave computes one 16x16 complex tile; K stepped by 4.
// cre/cim kept as cpos - cneg to avoid per-iteration operand negation.
__global__ void cgemm_wmma(const float* __restrict__ Are, const float* __restrict__ Aim,
                           const float* __restrict__ Bre, const float* __restrict__ Bim,
                           float* __restrict__ Dre, float* __restrict__ Dim,
                           int M, int N, int K,
                           int lda, int ldb,
                           long sA, long sB, long sD) {
    int wave = (int)((blockIdx.x * blockDim.x + threadIdx.x) >> 5);
    int lane = threadIdx.x & 31;
    int tilesN = N >> 4;
    int tilesPer = (M >> 4) * tilesN;
    int c  = wave / tilesPer;
    int t  = wave % tilesPer;
    int tm = t / tilesN, tn = t % tilesN;

    int lo = lane & 15;
    int hi = lane >> 4;
    int kb = hi << 1;                               // K-pair of this half-wave

    const float* ar = Are + (long)c * sA + (long)((tm << 4) + lo) * lda + kb;
    const float* ai = Aim + (long)c * sA + (long)((tm << 4) + lo) * lda + kb;
    const float* br = Bre + (long)c * sB + (long)((tn << 4) + lo) * ldb + kb;
    const float* bi = Bim + (long)c * sB + (long)((tn << 4) + lo) * ldb + kb;

    v8f cpos = {}; v8f cneg = {}; v8f cim = {};
    for (int k = 0; k < K; k += 4) {
        v2f Ar = *(const v2f*)(ar + k);
        v2f Ai = *(const v2f*)(ai + k);
        v2f Br = *(const v2f*)(br + k);
        v2f Bi = *(const v2f*)(bi + k);
        cpos = __builtin_amdgcn_wmma_f32_16x16x4_f32(false, Ar, false, Br, (short)0, cpos, false, false);
        cneg = __builtin_amdgcn_wmma_f32_16x16x4_f32(false, Ai, false, Bi, (short)0, cneg, false, false);
        cim  = __builtin_amdgcn_wmma_f32_16x16x4_f32(false, Ar, false, Bi, (short)0, cim,  false, false);
        cim  = __builtin_amdgcn_wmma_f32_16x16x4_f32(false, Ai, false, Br, (short)0, cim,  false, false);
    }

    // D layout: lane owns (n = tile_n + lo, m = tile_m + hi*8 + r), r = 0..7
    // -> 8 contiguous floats in the transposed plane: two b128 stores per plane.
    long dbase = (long)c * sD + (long)((tn << 4) + lo) * M + (tm << 4) + (hi << 3);
    float4* pre = (float4*)(Dre + dbase);
    float4* pim = (float4*)(Dim + dbase);
    pre[0] = make_float4(cpos[0] - cneg[0], cpos[1] - cneg[1], cpos[2] - cneg[2], cpos[3] - cneg[3]);
    pre[1] = make_float4(cpos[4] - cneg[4], cpos[5] - cneg[5], cpos[6] - cneg[6], cpos[7] - cneg[7]);
    pim[0] = make_float4(cim[0], cim[1], cim[2], cim[3]);
    pim[1] = make_float4(cim[4], cim[5], cim[6], cim[7]);
}

// ---------------------------------------------------------------- coil combine
// img planes are stored TRANSPOSED: img_t[c][n*N_IMG + m] for image pixel (m,n).
__global__ void combine(const float* __restrict__ img_re,
                        const float* __restrict__ img_im,
                        const float* __restrict__ smaps,  // (C,N,N,2)
                        float* __restrict__ out) {        // (N,N,2)
    int idx = blockIdx.x * blockDim.x + threadIdx.x;
    if (idx >= N_IMG * N_IMG) return;
    int m = idx >> 8, n = idx & (N_IMG - 1);
    int tidx = n * N_IMG + m;                       // transposed read
    float fm = ((float)m - 128.0f) / (float)G_DIM;
    float fn = ((float)n - 128.0f) / (float)G_DIM;
    float inv = 1.0f / (kb_ft(fm) * kb_ft(fn));

    float accr = 0.0f, acci = 0.0f;
#pragma unroll
    for (int c = 0; c < NCOIL; ++c) {
        float xr = img_re[(long)c * N_IMG * N_IMG + tidx];
        float xi = img_im[(long)c * N_IMG * N_IMG + tidx];
        const float* s = smaps + ((long)c * N_IMG * N_IMG + idx) * 2;
        float sr = s[0], si = s[1];
        accr += xr * sr + xi * si;                  // img * conj(smap)
        acci += xi * sr - xr * si;
    }
    out[idx * 2]     = accr * inv;
    out[idx * 2 + 1] = acci * inv;
}

// ---------------------------------------------------------------- launch
extern "C" void kernel_launch(void* const* d_in, const int* in_sizes, int n_in,
                              void* d_out, int out_size, void* d_ws, size_t ws_size,
                              hipStream_t stream) {
    const float* inp   = (const float*)d_in[0];
    const float* smaps = (const float*)d_in[1];
    const float* ktraj = (const float*)d_in[2];
    const float* dcomp = (const float*)d_in[3];
    float* out = (float*)d_out;
    const int K = in_sizes[3];                      // dcomp is (B,1,K)

    // workspace layout (floats), all split re/im planes
    const long GG  = (long)G_DIM * G_DIM;           // 262144
    const long CGG = (long)NCOIL * GG;              // 3,145,728
    const long CTT = (long)NCOIL * G_DIM * N_IMG;   // 1,572,864
    const long CII = (long)NCOIL * N_IMG * N_IMG;   //   786,432
    float* f = (float*)d_ws;
    float* gre  = f;             f += CGG;
    float* gim  = f;             f += CGG;
    float* Wre  = f;             f += GG;
    float* Wim  = f;             f += GG;
    float* Tre  = f;             f += CTT;          // stored transposed: [c][n*512 + k]
    float* Tim  = f;             f += CTT;
    float* Ire  = f;             f += CII;          // stored transposed: [c][n*256 + m]
    float* Iim  = f;             f += CII;

    // 1) zero the accumulation grid (every call; graph-replay determinism)
    zero_f32<<<2048, 256, 0, stream>>>(gre, (int)(2 * CGG));   // gre+gim contiguous
    // 2) DFT matrix
    fill_W<<<(G_DIM * G_DIM + 255) / 256, 256, 0, stream>>>(Wre, Wim);
    // 3) gridding
    gridding<<<(K + 255) / 256, 256, 0, stream>>>(inp, ktraj, dcomp, gre, gim, K);
    // 4) T[r][n] = sum_k grid[r][k] * W[k][n]  (B = W, symmetric -> [n*512+k])
    //    output written transposed into T_t[n*512 + r].  M=512, N=256, K=512.
    {
        int waves = NCOIL * (G_DIM / 16) * (N_IMG / 16);       // 6144
        cgemm_wmma<<<waves / 8, 256, 0, stream>>>(
            gre, gim, Wre, Wim, Tre, Tim,
            G_DIM, N_IMG, G_DIM,
            G_DIM, G_DIM,
            GG, 0L, (long)G_DIM * N_IMG);
    }
    // 5) img[m][n] = sum_k W[m][k] * T[k][n]  (B = T_t -> [n*512+k])
    //    output written transposed into img_t[n*256 + m].  M=256, N=256, K=512.
    {
        int waves = NCOIL * (N_IMG / 16) * (N_IMG / 16);       // 3072
        cgemm_wmma<<<waves / 8, 256, 0, stream>>>(
            Wre, Wim, Tre, Tim, Ire, Iim,
            N_IMG, N_IMG, G_DIM,
            G_DIM, G_DIM,
            0L, (long)G_DIM * N_IMG, (long)N_IMG * N_IMG);
    }
    // 6) apodization + coil combine
    combine<<<(N_IMG * N_IMG + 255) / 256, 256, 0, stream>>>(Ire, Iim, smaps, out);
}